// GCNs_26044681682965
// MI455X (gfx1250) — compile-verified
//
#include <hip/hip_runtime.h>

#define N_NODES   150000
#define N_EDGES   4800000
#define IN_DIM    1433
#define HID_DIM   16
#define OUT_DIM   7
#define OUT_STRIDE 8
#define N_TILES   (N_NODES / 16)   // 9375, exact

typedef __attribute__((ext_vector_type(2))) float v2f;
typedef __attribute__((ext_vector_type(8))) float v8f;

// ---------------------------------------------------------------- zero scratch
__global__ __launch_bounds__(256) void zero_kernel(float4* __restrict__ p, int n4) {
    int i = blockIdx.x * 256 + threadIdx.x;
    if (i < n4) p[i] = make_float4(0.f, 0.f, 0.f, 0.f);
}

// ---------------------------------------------------------------- GEMM1: xw1 = X @ W1
// One wave per 16-row tile. C tile = 16x16 f32 in 8 VGPRs.
// A frag (16x4 f32): lane L -> row L%16, K = k0 + 2*(L/16) + {0,1}
// B frag (4x16 f32): lane L -> col L%16, same K pairing
__global__ __launch_bounds__(256) void gemm1_kernel(const float* __restrict__ X,
                                                    const float* __restrict__ W1,
                                                    float* __restrict__ XW1) {
    const int wave = (blockIdx.x * 256 + threadIdx.x) >> 5;
    const int lane = threadIdx.x & 31;
    if (wave >= N_TILES) return;

    const int row0 = wave * 16;
    const int m  = lane & 15;          // row (A) / col (B) within tile
    const int kg = (lane >> 4) * 2;    // K sub-group: 0 or 2

    const float* xrow = X  + (size_t)(row0 + m) * IN_DIM + kg;   // A source
    const float* wcol = W1 + (size_t)kg * HID_DIM + m;           // B source

    v8f c = {};
    // 358 full K-steps cover k = 0..1431
    for (int k0 = 0; k0 + 4 <= IN_DIM - 1; k0 += 4) {
        v2f a, b;
        a.x = xrow[k0];
        a.y = xrow[k0 + 1];
        b.x = wcol[(size_t)k0 * HID_DIM];
        b.y = wcol[(size_t)(k0 + 1) * HID_DIM];
        c = __builtin_amdgcn_wmma_f32_16x16x4_f32(false, a, false, b,
                                                  (short)0, c, false, false);
    }
    // tail: k = 1432 only (lanes with kg==0, element 0); rest zero-padded
    {
        v2f a = {0.f, 0.f}, b = {0.f, 0.f};
        if (kg == 0) {
            a.x = xrow[IN_DIM - 1];
            b.x = wcol[(size_t)(IN_DIM - 1) * HID_DIM];
        }
        c = __builtin_amdgcn_wmma_f32_16x16x4_f32(false, a, false, b,
                                                  (short)0, c, false, false);
    }
    // C/D layout: VGPR r, lanes 0-15 -> M=r, lanes 16-31 -> M=r+8; N = lane%16
    float* o = XW1 + (size_t)(row0 + 8 * (lane >> 4)) * HID_DIM + m;
#pragma unroll
    for (int r = 0; r < 8; ++r) o[(size_t)r * HID_DIM] = c[r];
}

// ---------------------------------------------------------------- edge pass 1 (width 16)
__global__ __launch_bounds__(256) void edge1_kernel(const int* __restrict__ src,
                                                    const int* __restrict__ dst,
                                                    const float* __restrict__ vals,
                                                    const float* __restrict__ XW1,
                                                    float* __restrict__ AGG1) {
    int e = blockIdx.x * 256 + threadIdx.x;
    if (e >= N_EDGES) return;
    const int   s = src[e];
    const int   d = dst[e];
    const float a = vals[e];
    const float4* xr = (const float4*)(XW1 + (size_t)s * HID_DIM);
    float* o = AGG1 + (size_t)d * HID_DIM;
#pragma unroll
    for (int q = 0; q < 4; ++q) {
        float4 v = xr[q];
        atomicAdd(o + 4 * q + 0, a * v.x);
        atomicAdd(o + 4 * q + 1, a * v.y);
        atomicAdd(o + 4 * q + 2, a * v.z);
        atomicAdd(o + 4 * q + 3, a * v.w);
    }
}

// ---------------------------------------------------------------- GEMM2: xw2 = relu(agg1+b1) @ W2
__global__ __launch_bounds__(256) void gemm2_kernel(const float* __restrict__ AGG1,
                                                    const float* __restrict__ b1,
                                                    const float* __restrict__ W2,
                                                    float* __restrict__ XW2) {
    const int wave = (blockIdx.x * 256 + threadIdx.x) >> 5;
    const int lane = threadIdx.x & 31;
    if (wave >= N_TILES) return;

    const int row0 = wave * 16;
    const int m  = lane & 15;
    const int kg = (lane >> 4) * 2;

    const float* arow = AGG1 + (size_t)(row0 + m) * HID_DIM + kg;

    v8f c = {};
#pragma unroll
    for (int k0 = 0; k0 < HID_DIM; k0 += 4) {
        const int k = k0 + kg;
        v2f a, b;
        a.x = fmaxf(arow[k0]     + b1[k],     0.f);
        a.y = fmaxf(arow[k0 + 1] + b1[k + 1], 0.f);
        b.x = (m < OUT_DIM) ? W2[(size_t)k       * OUT_DIM + m] : 0.f;
        b.y = (m < OUT_DIM) ? W2[(size_t)(k + 1) * OUT_DIM + m] : 0.f;
        c = __builtin_amdgcn_wmma_f32_16x16x4_f32(false, a, false, b,
                                                  (short)0, c, false, false);
    }
    if (m < OUT_DIM) {
        float* o = XW2 + (size_t)(row0 + 8 * (lane >> 4)) * OUT_STRIDE + m;
#pragma unroll
        for (int r = 0; r < 8; ++r) o[(size_t)r * OUT_STRIDE] = c[r];
    }
}

// ---------------------------------------------------------------- edge pass 2 (width 7, stride 8)
__global__ __launch_bounds__(256) void edge2_kernel(const int* __restrict__ src,
                                                    const int* __restrict__ dst,
                                                    const float* __restrict__ vals,
                                                    const float* __restrict__ XW2,
                                                    float* __restrict__ AGG2) {
    int e = blockIdx.x * 256 + threadIdx.x;
    if (e >= N_EDGES) return;
    const int   s = src[e];
    const int   d = dst[e];
    const float a = vals[e];
    const float4* xr = (const float4*)(XW2 + (size_t)s * OUT_STRIDE);
    float4 v0 = xr[0];
    float4 v1 = xr[1];
    float* o = AGG2 + (size_t)d * OUT_STRIDE;
    atomicAdd(o + 0, a * v0.x);
    atomicAdd(o + 1, a * v0.y);
    atomicAdd(o + 2, a * v0.z);
    atomicAdd(o + 3, a * v0.w);
    atomicAdd(o + 4, a * v1.x);
    atomicAdd(o + 5, a * v1.y);
    atomicAdd(o + 6, a * v1.z);
}

// ---------------------------------------------------------------- log_softmax(agg2 + b2)
__global__ __launch_bounds__(256) void lsm_kernel(const float* __restrict__ AGG2,
                                                  const float* __restrict__ b2,
                                                  float* __restrict__ out) {
    int n = blockIdx.x * 256 + threadIdx.x;
    if (n >= N_NODES) return;
    const float* r = AGG2 + (size_t)n * OUT_STRIDE;
    float v[OUT_DIM];
    float mx = -INFINITY;
#pragma unroll
    for (int j = 0; j < OUT_DIM; ++j) {
        v[j] = r[j] + b2[j];
        mx = fmaxf(mx, v[j]);
    }
    float s = 0.f;
#pragma unroll
    for (int j = 0; j < OUT_DIM; ++j) s += expf(v[j] - mx);
    const float lse = mx + logf(s);
    float* o = out + (size_t)n * OUT_DIM;
#pragma unroll
    for (int j = 0; j < OUT_DIM; ++j) o[j] = v[j] - lse;
}

// ----------------------------------------------------------------
extern "C" void kernel_launch(void* const* d_in, const int* in_sizes, int n_in,
                              void* d_out, int out_size, void* d_ws, size_t ws_size,
                              hipStream_t stream) {
    const float* features = (const float*)d_in[0];
    const int*   esrc     = (const int*)  d_in[1];
    const int*   edst     = (const int*)  d_in[2];
    const float* avals    = (const float*)d_in[3];
    const float* W1       = (const float*)d_in[4];
    const float* b1       = (const float*)d_in[5];
    const float* W2       = (const float*)d_in[6];
    const float* b2       = (const float*)d_in[7];
    float* out = (float*)d_out;

    // workspace layout: [xw1 | agg1 | xw2 | agg2]
    float* xw1  = (float*)d_ws;
    float* agg1 = xw1  + (size_t)N_NODES * HID_DIM;
    float* xw2  = agg1 + (size_t)N_NODES * HID_DIM;
    float* agg2 = xw2  + (size_t)N_NODES * OUT_STRIDE;

    // zero agg1 .. end (agg1 + xw2 + agg2 = contiguous; xw2 overwritten anyway)
    {
        const int n4 = (N_NODES * (HID_DIM + OUT_STRIDE + OUT_STRIDE)) / 4;
        zero_kernel<<<(n4 + 255) / 256, 256, 0, stream>>>((float4*)agg1, n4);
    }

    const int gemm_blocks = (N_TILES * 32 + 255) / 256;
    gemm1_kernel<<<gemm_blocks, 256, 0, stream>>>(features, W1, xw1);

    edge1_kernel<<<(N_EDGES + 255) / 256, 256, 0, stream>>>(esrc, edst, avals, xw1, agg1);

    gemm2_kernel<<<gemm_blocks, 256, 0, stream>>>(agg1, b1, W2, xw2);

    edge2_kernel<<<(N_EDGES + 255) / 256, 256, 0, stream>>>(esrc, edst, avals, xw2, agg2);

    lsm_kernel<<<(N_NODES + 255) / 256, 256, 0, stream>>>(agg2, b2, out);
}